// GCNN_88991722373506
// MI455X (gfx1250) — compile-verified
//
#include <hip/hip_runtime.h>

// ---------------------------------------------------------------------------
// 2-layer GCN for MI455X (gfx1250).
// GEMMs use V_WMMA_F32_16X16X4_F32 (full-precision f32 WMMA); workload is
// scatter/gather bound (~1.7GB edge traffic @ 23.3TB/s dominates; the
// 3.3 GFLOP GEMM is noise) so no precision downcast is warranted.
// B-matrix is staged into LDS K-pair-swizzled so each WMMA B fragment is one
// aligned ds_load_b64 into an even VGPR pair (no v_mov repacking).
// ---------------------------------------------------------------------------

typedef __attribute__((ext_vector_type(2))) float v2f;
typedef __attribute__((ext_vector_type(8))) float v8f;

#define N_NODES 100000
#define F_IN    256
#define F_HID   64
#define F_OUT   4

// ----------------------------- degree / norm -------------------------------

__global__ void deg_init_kernel(float* __restrict__ deg) {
    int i = blockIdx.x * blockDim.x + threadIdx.x;
    if (i < N_NODES) deg[i] = 1.0f;   // self loop
}

__global__ void deg_accum_kernel(const int* __restrict__ dst, int E,
                                 float* __restrict__ deg) {
    int e = blockIdx.x * blockDim.x + threadIdx.x;
    if (e < E) atomicAdd(&deg[dst[e]], 1.0f);
}

__global__ void dinv_kernel(const float* __restrict__ deg,
                            float* __restrict__ dinv) {
    int i = blockIdx.x * blockDim.x + threadIdx.x;
    if (i < N_NODES) dinv[i] = rsqrtf(deg[i]);
}

__global__ void edge_norm_kernel(const int* __restrict__ src,
                                 const int* __restrict__ dst,
                                 const float* __restrict__ dinv,
                                 float* __restrict__ norm, int E) {
    int e = blockIdx.x * blockDim.x + threadIdx.x;
    if (e < E) norm[e] = dinv[src[e]] * dinv[dst[e]];
}

// ------------------------------- GEMM 1 ------------------------------------
// h1[100000,64] = x[100000,256] @ W1[256,64], f32 WMMA 16x16x4.
// One wave per 16-row tile; 4 accumulators cover all 64 output columns.
// A frag (16x4 f32): lanes 0-15 hold K=k,k+1; lanes 16-31 hold K=k+2,k+3
//   -> contiguous global_load_b64 from the row.
// B frag (4x16 f32): lane%16 = column; same K split across lane halves.
//   LDS swizzle: lw[((k>>1)*64 + col)*2 + (k&1)] so frag (k+2*hi, col) is one
//   aligned ds_load_b64 at pair index ((k>>1)+hi)*64 + col.
// C layout: VGPR i -> M = i + 8*(lane>=16); lane%16 = N.

__global__ void __launch_bounds__(256)
gemm1_wmma_kernel(const float* __restrict__ x,
                  const float* __restrict__ W1,
                  float* __restrict__ h1) {
    __shared__ float lw[F_IN * F_HID];            // 64 KB of 320 KB WGP LDS
    for (int i = threadIdx.x; i < F_IN * F_HID; i += blockDim.x) {
        int kk  = i >> 6;          // / F_HID
        int col = i & 63;          // % F_HID
        lw[(((kk >> 1) * F_HID) + col) * 2 + (kk & 1)] = W1[i];
    }
    __syncthreads();

    const int lane  = threadIdx.x & 31;
    const int wave  = threadIdx.x >> 5;
    const int mtile = blockIdx.x * 8 + wave;      // 16-row tile (6250 total, exact)
    if (mtile * 16 >= N_NODES) return;            // uniform per wave
    const int row0 = mtile * 16;

    const int r  = lane & 15;
    const int hi = lane >> 4;

    v8f acc0 = {}, acc1 = {}, acc2 = {}, acc3 = {};
    const float* xrow = x + (size_t)(row0 + r) * F_IN;
    const v2f*   wp   = (const v2f*)lw;

    for (int k = 0; k < F_IN; k += 4) {
        // unconditional speculative prefetch of the next cacheline of this row
        // (<=128B past row end; dropped silently if ever invalid)
        __builtin_prefetch(&xrow[k + 32], 0, 3);
        v2f a = *(const v2f*)&xrow[k + 2 * hi];   // global_load_b64
        const int pb = ((k >> 1) + hi) * F_HID;   // swizzled pair base
        v2f b0 = wp[pb + r +  0];                 // ds_load_b64 each
        v2f b1 = wp[pb + r + 16];
        v2f b2 = wp[pb + r + 32];
        v2f b3 = wp[pb + r + 48];
        acc0 = __builtin_amdgcn_wmma_f32_16x16x4_f32(false, a, false, b0, (short)0, acc0, false, false);
        acc1 = __builtin_amdgcn_wmma_f32_16x16x4_f32(false, a, false, b1, (short)0, acc1, false, false);
        acc2 = __builtin_amdgcn_wmma_f32_16x16x4_f32(false, a, false, b2, (short)0, acc2, false, false);
        acc3 = __builtin_amdgcn_wmma_f32_16x16x4_f32(false, a, false, b3, (short)0, acc3, false, false);
    }

    float* out = h1 + (size_t)row0 * F_HID;
    for (int i = 0; i < 8; i++) {
        int m = i + 8 * hi;
        out[(size_t)m * F_HID + (r +  0)] = acc0[i];
        out[(size_t)m * F_HID + (r + 16)] = acc1[i];
        out[(size_t)m * F_HID + (r + 32)] = acc2[i];
        out[(size_t)m * F_HID + (r + 48)] = acc3[i];
    }
}

// --------------------------- edge scatter, layer 1 -------------------------
// 64 threads per edge: coalesced 256B gather of h1[src,*] and 64 contiguous
// GLOBAL_ATOMIC_ADD_F32 into agg1[dst,*]. norm[e] broadcasts from cache.

__global__ void scatter1_kernel(const int* __restrict__ src,
                                const int* __restrict__ dst,
                                const float* __restrict__ norm,
                                const float* __restrict__ h1,
                                float* __restrict__ agg1, int E) {
    int t = blockIdx.x * blockDim.x + threadIdx.x;
    if (t >= E * F_HID) return;
    int e = t >> 6;           // /64
    int c = t & 63;
    float v = h1[(size_t)src[e] * F_HID + c] * norm[e];
    atomicAdd(&agg1[(size_t)dst[e] * F_HID + c], v);
}

// self-loop + bias + leaky_relu, written in place into agg1
__global__ void post1_kernel(const float* __restrict__ h1,
                             const float* __restrict__ deg,
                             const float* __restrict__ b1,
                             float* __restrict__ agg1) {
    int t = blockIdx.x * blockDim.x + threadIdx.x;
    if (t >= N_NODES * F_HID) return;
    int i = t >> 6;
    int c = t & 63;
    float v = agg1[t] + h1[t] * (1.0f / deg[i]) + b1[c];
    agg1[t] = (v > 0.0f) ? v : 0.01f * v;
}

// ------------------------------- GEMM 2 ------------------------------------
// h2[100000,4] = h1act[100000,64] @ W2[64,4]; W2 zero-padded to 64x16 and
// K-pair-swizzled in LDS; only cols 0..3 are stored.

__global__ void __launch_bounds__(256)
gemm2_wmma_kernel(const float* __restrict__ h,
                  const float* __restrict__ W2,
                  float* __restrict__ h2) {
    __shared__ float lw[F_HID * 16];
    for (int i = threadIdx.x; i < F_HID * 16; i += blockDim.x) {
        int kk = i >> 4, nn = i & 15;
        float v = (nn < F_OUT) ? W2[kk * F_OUT + nn] : 0.0f;
        lw[(((kk >> 1) * 16) + nn) * 2 + (kk & 1)] = v;
    }
    __syncthreads();

    const int lane  = threadIdx.x & 31;
    const int wave  = threadIdx.x >> 5;
    const int mtile = blockIdx.x * 8 + wave;
    if (mtile * 16 >= N_NODES) return;
    const int row0 = mtile * 16;

    const int r  = lane & 15;
    const int hi = lane >> 4;

    v8f acc = {};
    const float* hrow = h + (size_t)(row0 + r) * F_HID;
    const v2f*   wp   = (const v2f*)lw;
    for (int k = 0; k < F_HID; k += 4) {
        v2f a = *(const v2f*)&hrow[k + 2 * hi];
        v2f b = wp[((k >> 1) + hi) * 16 + r];
        acc = __builtin_amdgcn_wmma_f32_16x16x4_f32(false, a, false, b, (short)0, acc, false, false);
    }

    if (r < F_OUT) {
        float* out = h2 + (size_t)row0 * F_OUT;
        for (int i = 0; i < 8; i++) {
            int m = i + 8 * hi;
            out[(size_t)m * F_OUT + r] = acc[i];
        }
    }
}

// --------------------------- edge scatter, layer 2 -------------------------

__global__ void scatter2_kernel(const int* __restrict__ src,
                                const int* __restrict__ dst,
                                const float* __restrict__ norm,
                                const float* __restrict__ h2,
                                float* __restrict__ agg2, int E) {
    int e = blockIdx.x * blockDim.x + threadIdx.x;
    if (e >= E) return;
    float nrm = norm[e];
    const float4* hv = (const float4*)h2;
    float4 v = hv[src[e]];
    float* o = agg2 + (size_t)dst[e] * F_OUT;
    atomicAdd(&o[0], v.x * nrm);
    atomicAdd(&o[1], v.y * nrm);
    atomicAdd(&o[2], v.z * nrm);
    atomicAdd(&o[3], v.w * nrm);
}

// self-loop + bias + softmax over the 4 classes
__global__ void post2_kernel(const float* __restrict__ h2,
                             const float* __restrict__ deg,
                             const float* __restrict__ b2,
                             const float* __restrict__ agg2,
                             float* __restrict__ out) {
    int i = blockIdx.x * blockDim.x + threadIdx.x;
    if (i >= N_NODES) return;
    float inv = 1.0f / deg[i];
    float v[F_OUT];
    float mx = -3.4e38f;
    for (int c = 0; c < F_OUT; c++) {
        v[c] = agg2[(size_t)i * F_OUT + c] + h2[(size_t)i * F_OUT + c] * inv + b2[c];
        mx = fmaxf(mx, v[c]);
    }
    float s = 0.0f;
    for (int c = 0; c < F_OUT; c++) { v[c] = __expf(v[c] - mx); s += v[c]; }
    float rs = 1.0f / s;
    for (int c = 0; c < F_OUT; c++) out[(size_t)i * F_OUT + c] = v[c] * rs;
}

// ---------------------------------------------------------------------------

extern "C" void kernel_launch(void* const* d_in, const int* in_sizes, int n_in,
                              void* d_out, int out_size, void* d_ws, size_t ws_size,
                              hipStream_t stream) {
    const float* x  = (const float*)d_in[0];
    const int*   ei = (const int*)d_in[1];      // edge_index [2,E] (int per harness)
    const float* W1 = (const float*)d_in[2];
    const float* b1 = (const float*)d_in[3];
    const float* W2 = (const float*)d_in[4];
    const float* b2 = (const float*)d_in[5];
    float* out = (float*)d_out;

    const int E = in_sizes[1] / 2;
    const int* src = ei;
    const int* dst = ei + E;

    // workspace layout (floats): deg[N] dinv[N] norm[E] h1[N*64] agg1[N*64]
    //                            h2[N*4] agg2[N*4]   (~68 MB total)
    float* ws   = (float*)d_ws;
    float* deg  = ws;
    float* dinv = deg  + N_NODES;
    float* norm = dinv + N_NODES;
    float* h1   = norm + E;
    float* agg1 = h1   + (size_t)N_NODES * F_HID;
    float* h2   = agg1 + (size_t)N_NODES * F_HID;
    float* agg2 = h2   + (size_t)N_NODES * F_OUT;

    const int T = 256;
    const int nb_nodes  = (N_NODES + T - 1) / T;
    const int nb_edges  = (E + T - 1) / T;
    const int nb_tiles  = (N_NODES / 16 + 7) / 8;            // 782 blocks, 8 waves each
    const int nb_nf     = (N_NODES * F_HID + T - 1) / T;
    const long long s1t = (long long)E * F_HID;
    const int nb_s1     = (int)((s1t + T - 1) / T);

    // degrees + per-edge normalization
    deg_init_kernel<<<nb_nodes, T, 0, stream>>>(deg);
    deg_accum_kernel<<<nb_edges, T, 0, stream>>>(dst, E, deg);
    dinv_kernel<<<nb_nodes, T, 0, stream>>>(deg, dinv);
    edge_norm_kernel<<<nb_edges, T, 0, stream>>>(src, dst, dinv, norm, E);

    // layer 1
    gemm1_wmma_kernel<<<nb_tiles, T, 0, stream>>>(x, W1, h1);
    hipMemsetAsync(agg1, 0, (size_t)N_NODES * F_HID * sizeof(float), stream);
    scatter1_kernel<<<nb_s1, T, 0, stream>>>(src, dst, norm, h1, agg1, E);
    post1_kernel<<<nb_nf, T, 0, stream>>>(h1, deg, b1, agg1);

    // layer 2
    gemm2_wmma_kernel<<<nb_tiles, T, 0, stream>>>(agg1, W2, h2);
    hipMemsetAsync(agg2, 0, (size_t)N_NODES * F_OUT * sizeof(float), stream);
    scatter2_kernel<<<nb_edges, T, 0, stream>>>(src, dst, norm, h2, agg2, E);
    post2_kernel<<<nb_nodes, T, 0, stream>>>(h2, deg, b2, agg2, out);
}